// Kernel_19164144075225
// MI455X (gfx1250) — compile-verified
//
#include <hip/hip_runtime.h>
#include <math.h>

typedef __attribute__((ext_vector_type(2))) float v2f;
typedef __attribute__((ext_vector_type(8))) float v8f;

constexpr int B = 16;
constexpr int N = 2048;

#define PERIM    6.2831853071795864769f
#define INVPERIM 0.15915494309189533577f
#define HALFP    3.1415926535897932385f
#define SCALE2LG 2.8853900817779268147f   // 2 * log2(e)

// x mod 2pi with result in [0, 2pi) -- cheap, no libm fmod NaN chain.
__device__ __forceinline__ float pmod2pi(float x) {
    return x - PERIM * floorf(x * INVPERIM);
}

// Raw v_max_num_f32: avoids llvm.maxnum's canonicalize(max(x,x)) on each input.
__device__ __forceinline__ float maxnum(float a, float b) {
    float r;
    asm("v_max_num_f32 %0, %1, %2" : "=v"(r) : "v"(a), "v"(b));
    return r;
}

// Single v_exp_f32 (hardware computes 2^x).
__device__ __forceinline__ float exp2_hw(float x) {
    return __builtin_amdgcn_exp2f(x);
}

// f32 16x16x4 operand packing (ISA 7.12.2):
//   VGPR0 = K0 (lanes 0-15) | K2 (lanes 16-31)
//   VGPR1 = K1 (lanes 0-15) | K3 (lanes 16-31)
// A row m: 2*log2e * [eta, pm, sq, 1]    B col n: [eta, pm, -1/2, -sq/2]
// => D[m][n] = 2*log2e * (dot - (sq_m+sq_n)/2) = -log2e * d
// => adj = exp(-d) = exp2(max(D1, D2))      (single v_exp_f32, no pre-mul)
__device__ __forceinline__ v2f packA(bool lo, float e, float pm, float sq) {
    v2f a;
    a.x = SCALE2LG * (lo ? e  : sq);
    a.y = lo ? (SCALE2LG * pm) : SCALE2LG;
    return a;
}
__device__ __forceinline__ v2f packB(bool lo, float e, float pm, float sq) {
    v2f b;
    b.x = lo ? e  : -0.5f;
    b.y = lo ? pm : -0.5f * sq;
    return b;
}

// ---------------------------------------------------------------------------
// Pass 1: degrees. One wave owns a 16-row strip; loops over 128 column tiles.
// Two V_WMMA_F32_16X16X4_F32 per tile (phi and phi+pi variants), norms and
// exp-scaling folded into the K=2,3 slots; row sums accumulated in VGPRs.
// ---------------------------------------------------------------------------
__global__ void degree_kernel(const float* __restrict__ phi,
                              const float* __restrict__ eta,
                              float* __restrict__ degree) {
    const int lane = threadIdx.x & 31;
    const int wave = threadIdx.x >> 5;
    const int strip = blockIdx.x * 8 + wave;          // [0, B*N/16)
    const int b  = strip >> 7;                        // / (N/16)
    const int i0 = (strip & 127) << 4;                // * 16
    const int l15 = lane & 15;
    const bool lo = lane < 16;

    // Row point for row (i0 + l15); loop-invariant A operands (pre-scaled).
    float re = eta[b * N + i0 + l15];
    float rp = phi[b * N + i0 + l15];
    float rp1 = pmod2pi(rp);
    float rp2 = pmod2pi(rp + HALFP);
    float rsq1 = re * re + rp1 * rp1;
    float rsq2 = re * re + rp2 * rp2;
    v2f a1 = packA(lo, re, rp1, rsq1);
    v2f a2 = packA(lo, re, rp2, rsq2);

    float acc[8] = {0, 0, 0, 0, 0, 0, 0, 0};

    for (int j0 = 0; j0 < N; j0 += 16) {
        float ce = eta[b * N + j0 + l15];
        float cp = phi[b * N + j0 + l15];
        float cp1 = pmod2pi(cp);
        float cp2 = pmod2pi(cp + HALFP);
        float csq1 = ce * ce + cp1 * cp1;
        float csq2 = ce * ce + cp2 * cp2;
        v2f b1 = packB(lo, ce, cp1, csq1);
        v2f b2 = packB(lo, ce, cp2, csq2);

        v8f z = {};
        v8f c1 = __builtin_amdgcn_wmma_f32_16x16x4_f32(
            false, a1, false, b1, (short)0, z, false, false);
        v8f c2 = __builtin_amdgcn_wmma_f32_16x16x4_f32(
            false, a2, false, b2, (short)0, z, false, false);

#pragma unroll
        for (int r = 0; r < 8; ++r) {
            acc[r] += exp2_hw(maxnum(c1[r], c2[r]));
        }
    }

    // Reduce across the 16 columns held within each half-wave.
#pragma unroll
    for (int r = 0; r < 8; ++r) {
        float v = acc[r];
        v += __shfl_xor(v, 1, 32);
        v += __shfl_xor(v, 2, 32);
        v += __shfl_xor(v, 4, 32);
        v += __shfl_xor(v, 8, 32);
        if (l15 == r) {
            int m = lo ? r : (r + 8);
            degree[b * N + i0 + m] = v;
        }
    }
}

// ---------------------------------------------------------------------------
// Pass 2: per-batch bitonic argsort (descending degree, index tiebreak) in LDS,
// plus the permuted per-node vectors.
// ---------------------------------------------------------------------------
__global__ void sort_kernel(const float* __restrict__ degree,
                            const float* __restrict__ e,
                            const float* __restrict__ phi,
                            const float* __restrict__ eta,
                            int*  __restrict__ order,
                            float* __restrict__ e_o,
                            float* __restrict__ phi_o,
                            float* __restrict__ eta_o) {
    __shared__ float key[N];
    __shared__ int   val[N];
    const int b = blockIdx.x;
    const int tid = threadIdx.x;

    for (int i = tid; i < N; i += 1024) { key[i] = degree[b * N + i]; val[i] = i; }
    __syncthreads();

    for (int k = 2; k <= N; k <<= 1) {
        for (int j = k >> 1; j > 0; j >>= 1) {
            for (int i = tid; i < N; i += 1024) {
                int ixj = i ^ j;
                if (ixj > i) {
                    float ka = key[i], kb = key[ixj];
                    int   va = val[i], vb = val[ixj];
                    bool aFirst = (ka > kb) || (ka == kb && va < vb); // desc, stable-ish
                    bool up = ((i & k) == 0);
                    if (up ? !aFirst : aFirst) {
                        key[i] = kb; key[ixj] = ka;
                        val[i] = vb; val[ixj] = va;
                    }
                }
            }
            __syncthreads();
        }
    }

    for (int i = tid; i < N; i += 1024) {
        int o = val[i];
        order[b * N + i] = o;
        e_o  [b * N + i] = e  [b * N + o];
        phi_o[b * N + i] = phi[b * N + o];
        eta_o[b * N + i] = eta[b * N + o];
    }
}

// ---------------------------------------------------------------------------
// Pass 3: adj_o recomputed directly at permuted positions (no 268MB round-trip
// through an unpermuted adj). One wave = one permuted 16x16 tile: gather node
// ids via order[], two WMMAs, exp2, coalesced 16-float row stores.
// ---------------------------------------------------------------------------
__global__ void adj_kernel(const float* __restrict__ phi,
                           const float* __restrict__ eta,
                           const int*  __restrict__ order,
                           float* __restrict__ adj_o) {
    const int lane = threadIdx.x & 31;
    const int wave = threadIdx.x >> 5;
    const int l15 = lane & 15;
    const bool lo = lane < 16;
    const int jt = blockIdx.x;                 // column tile  [0, N/16)
    const int it = blockIdx.y * 8 + wave;      // row tile     [0, N/16)
    const int b  = blockIdx.z;

    const int ri = order[b * N + it * 16 + l15];
    const int ci = order[b * N + jt * 16 + l15];

    float re = eta[b * N + ri], rp = phi[b * N + ri];
    float rp1 = pmod2pi(rp), rp2 = pmod2pi(rp + HALFP);
    float rsq1 = re * re + rp1 * rp1;
    float rsq2 = re * re + rp2 * rp2;
    v2f a1 = packA(lo, re, rp1, rsq1);
    v2f a2 = packA(lo, re, rp2, rsq2);

    float ce = eta[b * N + ci], cp = phi[b * N + ci];
    float cp1 = pmod2pi(cp), cp2 = pmod2pi(cp + HALFP);
    float csq1 = ce * ce + cp1 * cp1;
    float csq2 = ce * ce + cp2 * cp2;
    v2f b1 = packB(lo, ce, cp1, csq1);
    v2f b2 = packB(lo, ce, cp2, csq2);

    v8f z = {};
    v8f c1 = __builtin_amdgcn_wmma_f32_16x16x4_f32(
        false, a1, false, b1, (short)0, z, false, false);
    v8f c2 = __builtin_amdgcn_wmma_f32_16x16x4_f32(
        false, a2, false, b2, (short)0, z, false, false);

    size_t base = (size_t)b * N * N + (size_t)(jt * 16 + l15);
#pragma unroll
    for (int r = 0; r < 8; ++r) {
        int m = lo ? r : (r + 8);
        adj_o[base + (size_t)(it * 16 + m) * N] =
            exp2_hw(maxnum(c1[r], c2[r]));
    }
}

extern "C" void kernel_launch(void* const* d_in, const int* in_sizes, int n_in,
                              void* d_out, int out_size, void* d_ws, size_t ws_size,
                              hipStream_t stream) {
    const float* e   = (const float*)d_in[0];
    const float* phi = (const float*)d_in[1];
    const float* eta = (const float*)d_in[2];

    float* adj_o = (float*)d_out;                       // [B, N, N]
    float* e_o   = adj_o + (size_t)B * N * N;           // [B, N]
    float* phi_o = e_o + (size_t)B * N;
    float* eta_o = phi_o + (size_t)B * N;

    float* degree = (float*)d_ws;                         // B*N floats
    int*   order  = (int*)((float*)d_ws + (size_t)B * N); // B*N ints

    degree_kernel<<<dim3(B * (N / 16) / 8), 256, 0, stream>>>(phi, eta, degree);
    sort_kernel<<<dim3(B), 1024, 0, stream>>>(degree, e, phi, eta, order,
                                              e_o, phi_o, eta_o);
    adj_kernel<<<dim3(N / 16, N / 128, B), 256, 0, stream>>>(phi, eta, order, adj_o);
}